// LSTM_3813930958978
// MI455X (gfx1250) — compile-verified
//
#include <hip/hip_runtime.h>
#include <hip/hip_bf16.h>

// ---------------------------------------------------------------------------
// LSTM on MI455X (gfx1250):
//   per-step fused GEMM  gates = [x_t | h] @ [W_ih | W_hh]^T + b_ih + b_hh
//   - V_WMMA_F32_16X16X4_F32 (full fp32 fidelity vs fp32 reference)
//   - GLOBAL_LOAD_ASYNC_TO_LDS_B128 double-buffered pipeline (ASYNCcnt),
//     overlapping tile k+1 DMA with tile k WMMAs
//   - weights (32 MB) stay L2-resident (192 MB L2) across all 512 steps;
//     x_t tiles are loaded with th:TH_LOAD_NT so the one-pass 128 MB x
//     stream does not evict them.
// ---------------------------------------------------------------------------

typedef float v2f  __attribute__((ext_vector_type(2)));
typedef float v8f  __attribute__((ext_vector_type(8)));

#define B_DIM   64
#define H_DIM   1024
#define G_DIM   4096          // 4*H
#define K_DIM   1024          // I == H
#define TILE_N  64
#define TILE_K  32
#define LDT     36            // padded row stride (floats) for As/Bs tiles:
                              //  - 36*4=144 B rows -> every (row,kq) dest is 16B aligned
                              //  - 36*dLane ≡ 2 (mod 64) unsolvable -> no lo/hi half-wave
                              //    bank collision; 9*lane mod 16 bijective -> no intra-half

// one 16B-per-lane async DMA: copies 16 bytes from global address gaddr to LDS offset lds_off
__device__ __forceinline__ void async_b128(unsigned lds_off, const float* gaddr)
{
    asm volatile("global_load_async_to_lds_b128 %0, %1, off"
                 :: "v"(lds_off), "v"(gaddr) : "memory");
}
__device__ __forceinline__ void async_b128_nt(unsigned lds_off, const float* gaddr)
{
    asm volatile("global_load_async_to_lds_b128 %0, %1, off th:TH_LOAD_NT"
                 :: "v"(lds_off), "v"(gaddr) : "memory");
}

// stage a 64-row x 32-col fp32 tile (row stride of src = 1024) into LDS,
// 2 async b128 instructions per thread (512 x 16B total)
template <bool NT>
__device__ __forceinline__ void load_tile_async(float* lds_base, const float* src, int k0)
{
    const int tid = threadIdx.x;
    #pragma unroll
    for (int rep = 0; rep < 2; ++rep) {
        const int e  = tid + rep * 256;      // 0..511
        const int r  = e >> 3;               // 0..63
        const int kq = (e & 7) << 2;         // 0,4,...,28
        const unsigned dst = (unsigned)(size_t)(lds_base + r * LDT + kq);
        const float* g = src + (size_t)r * K_DIM + k0 + kq;
        if (NT) async_b128_nt(dst, g);
        else    async_b128(dst, g);
    }
}

// gates[64,4096] = x_t @ W_ih^T + h @ W_hh^T + b_ih + b_hh
__global__ __launch_bounds__(256)
void lstm_gates_gemm(const float* __restrict__ xt,    // [64,1024]
                     const float* __restrict__ h,     // [64,1024]
                     const float* __restrict__ Wih,   // [4096,1024] row-major
                     const float* __restrict__ Whh,   // [4096,1024] row-major
                     const float* __restrict__ bih,   // [4096]
                     const float* __restrict__ bhh,   // [4096]
                     float* __restrict__ gates)       // [64,4096]
{
    __shared__ __align__(16) float As[2][B_DIM * LDT];   // activations tile (dbl buf)
    __shared__ __align__(16) float Bs[2][TILE_N * LDT];  // weight tile, row-major [n][k]

    const int tid   = threadIdx.x;
    const int lane  = tid & 31;
    const int wave  = tid >> 5;          // 0..7
    const int waveM = wave & 3;          // 4 row tiles of 16
    const int waveN = wave >> 2;         // 2 col tiles of 32
    const int hl    = lane >> 4;         // half-wave: 0 -> K{0,1}, 1 -> K{2,3}
    const int lm    = lane & 15;

    const int n0 = blockIdx.x * TILE_N;  // 64 blocks cover N=4096
    const float* WihN = Wih + (size_t)n0 * K_DIM;
    const float* WhhN = Whh + (size_t)n0 * K_DIM;

    v8f acc0 = {0.f,0.f,0.f,0.f,0.f,0.f,0.f,0.f};
    v8f acc1 = acc0;

    const int aRow  = waveM * 16 + lm;   // row index into As tile
    const int bRow0 = waveN * 32 + lm;   // row (=n) index into Bs tile
    const int bRow1 = bRow0 + 16;

    // ---- prologue: DMA chunk 0 (phase 0 = x_t, k0 = 0) into buffer 0 ----
    load_tile_async<true >(&As[0][0], xt,   0);
    load_tile_async<false>(&Bs[0][0], WihN, 0);

    // 64 K-chunks: cc in [0,32) -> A=x_t/W_ih, cc in [32,64) -> A=h/W_hh
    for (int cc = 0; cc < 64; ++cc) {
        const int cur = cc & 1;

        if (cc + 1 < 64) {
            const int nc  = cc + 1;
            const int nk0 = (nc & 31) * TILE_K;
            if (nc & 32) {
                load_tile_async<false>(&As[nc & 1][0], h,    nk0);
                load_tile_async<false>(&Bs[nc & 1][0], WhhN, nk0);
            } else {
                load_tile_async<true >(&As[nc & 1][0], xt,   nk0);
                load_tile_async<false>(&Bs[nc & 1][0], WihN, nk0);
            }
            // 4 newer async ops outstanding -> previous chunk's 4 are complete
            asm volatile("s_wait_asynccnt 0x4" ::: "memory");
        } else {
            asm volatile("s_wait_asynccnt 0x0" ::: "memory");
        }
        __syncthreads();   // all waves' DMA for buffer `cur` has landed

        const float* __restrict__ a_ = &As[cur][0];
        const float* __restrict__ b_ = &Bs[cur][0];

        // 8 WMMA K-steps of 4 over the 32-deep tile.
        // A frag (16x4): lanes 0-15 hold (M=lm, K{ks,ks+1}), lanes 16-31 K{+2,+3}.
        // B frag (4x16): VGPR0 = rows K0(lo)/K2(hi), VGPR1 = K1(lo)/K3(hi);
        //   gathered from row-major Bs[n][k] with conflict-free stride LDT.
        #pragma unroll
        for (int kk = 0; kk < TILE_K; kk += 4) {
            const int ks = kk + hl * 2;
            v2f a, b0, b1;
            a.x  = a_[aRow  * LDT + ks];
            a.y  = a_[aRow  * LDT + ks + 1];
            b0.x = b_[bRow0 * LDT + ks];
            b0.y = b_[bRow0 * LDT + ks + 1];
            b1.x = b_[bRow1 * LDT + ks];
            b1.y = b_[bRow1 * LDT + ks + 1];
            acc0 = __builtin_amdgcn_wmma_f32_16x16x4_f32(
                       false, a, false, b0, (short)0, acc0, false, false);
            acc1 = __builtin_amdgcn_wmma_f32_16x16x4_f32(
                       false, a, false, b1, (short)0, acc1, false, false);
        }
        __syncthreads();   // nobody may DMA over `cur^1`'s successor while reads pend
    }

    // ---- epilogue: C/D layout: VGPR r -> (M = waveM*16 + hl*8 + r, N = lm) ----
    const int gcol0 = n0 + waveN * 32 + lm;
    const int gcol1 = gcol0 + 16;
    const float bias0 = bih[gcol0] + bhh[gcol0];
    const float bias1 = bih[gcol1] + bhh[gcol1];
    #pragma unroll
    for (int r = 0; r < 8; ++r) {
        const int row = waveM * 16 + hl * 8 + r;
        gates[(size_t)row * G_DIM + gcol0] = acc0[r] + bias0;
        gates[(size_t)row * G_DIM + gcol1] = acc1[r] + bias1;
    }
}

// gate order i, f, g, o (torch LSTMCell): c' = sig(f)*c + sig(i)*tanh(g); h' = sig(o)*tanh(c')
__global__ __launch_bounds__(256)
void lstm_pointwise(const float* __restrict__ gates,  // [64,4096]
                    float* __restrict__ c,            // [64,1024]
                    float* __restrict__ h)            // [64,1024]
{
    const int idx = blockIdx.x * 256 + threadIdx.x;   // 0..65535
    const int b = idx >> 10;
    const int j = idx & (H_DIM - 1);
    const float* g = gates + (size_t)b * G_DIM;
    float gi = g[j];
    float gf = g[H_DIM + j];
    float gg = g[2 * H_DIM + j];
    float go = g[3 * H_DIM + j];
    gi = 1.f / (1.f + __expf(-gi));
    gf = 1.f / (1.f + __expf(-gf));
    go = 1.f / (1.f + __expf(-go));
    gg = tanhf(gg);
    const float cn = gf * c[idx] + gi * gg;
    c[idx] = cn;
    h[idx] = go * tanhf(cn);
}

__global__ __launch_bounds__(256)
void lstm_init(float* __restrict__ h, float* __restrict__ c)
{
    const int idx = blockIdx.x * 256 + threadIdx.x;   // 0..65535
    h[idx] = 0.f;
    c[idx] = 0.f;
}

extern "C" void kernel_launch(void* const* d_in, const int* in_sizes, int n_in,
                              void* d_out, int out_size, void* d_ws, size_t ws_size,
                              hipStream_t stream)
{
    (void)in_sizes; (void)n_in; (void)out_size; (void)ws_size;
    const float* x   = (const float*)d_in[0];   // [512,64,1024]
    const float* Wih = (const float*)d_in[1];   // [4096,1024]
    const float* Whh = (const float*)d_in[2];   // [4096,1024]
    const float* bih = (const float*)d_in[3];   // [4096]
    const float* bhh = (const float*)d_in[4];   // [4096]

    float* h     = (float*)d_out;                  // [64,1024] -- h lives in d_out
    float* gates = (float*)d_ws;                   // [64,4096]
    float* c     = gates + (size_t)B_DIM * G_DIM;  // [64,1024]

    lstm_init<<<256, 256, 0, stream>>>(h, c);
    for (int t = 0; t < 512; ++t) {
        lstm_gates_gemm<<<G_DIM / TILE_N, 256, 0, stream>>>(
            x + (size_t)t * B_DIM * K_DIM, h, Wih, Whh, bih, bhh, gates);
        lstm_pointwise<<<(B_DIM * H_DIM) / 256, 256, 0, stream>>>(gates, c, h);
    }
}